// DCN_sep_pre_multi_offset_flow_similarity_v2_1_41085657154045
// MI455X (gfx1250) — compile-verified
//
#include <hip/hip_runtime.h>
#include <hip/hip_bf16.h>
#include <math.h>

typedef __attribute__((ext_vector_type(2))) float v2f;
typedef __attribute__((ext_vector_type(8))) float v8f;

#define B_    2
#define C_    64
#define H_    128
#define W_    128
#define O_    64
#define DG_   8
#define KK_   9
#define CG_   (C_ / DG_)        // 8 channels per deformable group
#define KC1   (C_ * KK_)        // 576 = GEMM reduction dim (im2col)
#define OM_   (3 * DG_ * KK_)   // 216 conv_offset_mask out channels
#define OMP_  224               // zero-padded to 14 tiles of 16
#define OFFC  (2 * DG_ * KK_)   // 144 offset channels (= 9 full tiles of 16)
#define MSKC  (DG_ * KK_)       // 72 mask channels
#define HW_   (H_ * W_)
#define MRM_  10.0f

// LDS strides chosen for conflict-free wave32 access on 64 banks
#define WSTR  580               // weight row stride: A ds_load_b64 hits all 64 banks
#define BSTR  72                // im2col row stride (K1): K+2 => +144 dw = +16 banks

// K1 LDS partition (floats): patch | bsh | wsh
#define K1_PATCH   (C_ * 3 * 66)          // 12672: 64ch x 3 rows x 66 cols halo
#define K1_BSH     (KC1 * BSTR)           // 41472: im2col 576 x 64px (stride 72)
#define K1_WSH     (16 * WSTR)            // 9280:  one 16x576 weight tile
#define K1_SMEM    (K1_PATCH + K1_BSH + K1_WSH)   // 63424 floats = 253,696 B

// K2 LDS partition (floats): wsh2 | val
#define K2_WSH     (O_ * WSTR)            // 37120: full 64x576 weight matrix
#define K2_VAL     (KC1 * 16)             // 9216:  im2col 576 x 16px
#define K2_SMEM    (K2_WSH + K2_VAL)      // 46336 floats = 185,344 B

// ---------------------------------------------------------------------------
// Fast activations: hardware v_tanh_f32 when available; short branchless
// __expf-based fallbacks (v_exp_f32) otherwise. ~1-2 ulp, ample for the
// 10*tanh offset / sigmoid mask feeding bilinear sampling.
// ---------------------------------------------------------------------------
__device__ __forceinline__ float fast_tanh(float v) {
#if __has_builtin(__builtin_amdgcn_tanhf)
  return __builtin_amdgcn_tanhf(v);
#else
  float av = fabsf(v);
  float e  = __expf(-2.0f * av);
  float r  = __fdividef(1.0f - e, 1.0f + e);
  return copysignf(r, v);
#endif
}
__device__ __forceinline__ float fast_sigmoid(float s) {
  return __fdividef(1.0f, 1.0f + __expf(-s));
}

// ---------------------------------------------------------------------------
// Prep: zero-pad conv_offset_mask weights from 216 to 224 rows so all A-tile
// loads in kernel 1 are unconditional.
// ---------------------------------------------------------------------------
__global__ __launch_bounds__(256) void prep_wpad_kernel(
    const float* __restrict__ w_om, float* __restrict__ w_pad)
{
  int i = blockIdx.x * 256 + threadIdx.x;
  if (i < OMP_ * KC1)
    w_pad[i] = (i < OM_ * KC1) ? w_om[i] : 0.0f;   // row-major: oc<216 <=> i<216*576
}

// ---------------------------------------------------------------------------
// Kernel 1: conv_offset_mask as implicit GEMM (fp32 WMMA 16x16x4), fused with
//   offset = 10*tanh(om) + pre_offset   and   mask = sigmoid(om * pre_sim).
// One workgroup (4 waves) per 64-pixel half-row:
//   stage x1 halo patch -> expand linear im2col bsh[576][64] in LDS,
//   then for each of 14 M-tiles: stage 16x576 weights in LDS (shared by all
//   4 waves; each wave owns one 16-pixel N-tile) and run a pure-linear K-loop.
// ---------------------------------------------------------------------------
__global__ __launch_bounds__(128) void conv_offset_mask_kernel(
    const float* __restrict__ x1, const float* __restrict__ w_pad,
    const float* __restrict__ b_om, const float* __restrict__ pre_offset,
    const float* __restrict__ pre_sim, float* __restrict__ offs_ws,
    float* __restrict__ mask_ws)
{
  extern __shared__ float smem[];
  float* patch = smem;                       // [64][3][66]
  float* bsh   = smem + K1_PATCH;            // [576][BSTR]
  float* wsh   = smem + K1_PATCH + K1_BSH;   // [16][WSTR]

  const int wg = blockIdx.x;                 // B*H*2 = 512 workgroups
  const int b  = wg / (H_ * 2);
  const int r0 = wg % (H_ * 2);
  const int h  = r0 >> 1;
  const int wb = (r0 & 1) * 64;

  // ---- stage 1: halo patch of x1 (rows h-1..h+1, cols wb-1..wb+64), branchless
  for (int i = threadIdx.x; i < K1_PATCH; i += 128) {
    int c   = i / (3 * 66);
    int rem = i % (3 * 66);
    int row = rem / 66;
    int col = rem % 66;
    int hh = h - 1 + row;
    int ww = wb - 1 + col;
    float inb = (hh >= 0 && hh < H_ && ww >= 0 && ww < W_) ? 1.0f : 0.0f;
    int hc = min(max(hh, 0), H_ - 1);
    int wc = min(max(ww, 0), W_ - 1);
    patch[i] = inb * x1[((b * C_ + c) * H_ + hc) * W_ + wc];
  }
  __syncthreads();

  // ---- stage 2: expand to linear im2col  bsh[k = c*9+t][n], power-of-2 decode
  for (int s = threadIdx.x; s < KK_ * 64; s += 128) {
    int t = s >> 6;                  // tap 0..8
    int n = s & 63;                  // pixel 0..63
    int ky = t / 3, kx = t % 3;
    const float* prow = patch + ky * 66 + n + kx;
    float* brow = bsh + t * BSTR + n;
    #pragma unroll 8
    for (int c = 0; c < C_; ++c)
      brow[c * KK_ * BSTR] = prow[c * 3 * 66];
  }
  __syncthreads();

  const int lane = threadIdx.x & 31;
  const int wave = threadIdx.x >> 5;     // N-tile 0..3 (16 pixels each)
  const int m    = lane & 15;            // A: M row / B & D: N column
  const int half = lane >> 4;            // K pair {0,1} vs {2,3}
  const int nb   = wave * 16 + m;        // pixel within half-row

  for (int tile = 0; tile < 14; ++tile) {
    // stage this M-tile's weights via b128 (row bases are 16B aligned)
    #pragma unroll
    for (int r = 0; r < 16; ++r) {
      const float4* srcw = reinterpret_cast<const float4*>(
          w_pad + (size_t)(tile * 16 + r) * KC1);
      float4* dstw = reinterpret_cast<float4*>(wsh + r * WSTR);
      for (int q = threadIdx.x; q < KC1 / 4; q += 128)
        dstw[q] = srcw[q];
    }
    __syncthreads();

    v8f acc = {};
    const float* arow = wsh + m * WSTR;          // even base -> 8B aligned
    #pragma unroll 4
    for (int k4 = 0; k4 < KC1; k4 += 4) {
      const int ka = k4 + half * 2;
      v2f a = *(const v2f*)(arow + ka);          // ds_load_b64, 64-bank clean
      v2f bm;
      bm.x = bsh[ka * BSTR + nb];
      bm.y = bsh[(ka + 1) * BSTR + nb];
      acc = __builtin_amdgcn_wmma_f32_16x16x4_f32(
          false, a, false, bm, (short)0, acc, false, false);
    }

    // D layout: vgpr r -> (M = r + 8*half, N = m); fused activations.
    // OFFC = 144 = 9 tiles, so the offset/mask split is UNIFORM in `tile`;
    // tiles 9..12 are fully-valid mask tiles; in tile 13 validity == (half==0).
    const int p = h * W_ + wb + nb;
    if (tile < 9) {                      // scalar branch: all-offset tiles
      #pragma unroll
      for (int r = 0; r < 8; ++r) {
        int ocr = tile * 16 + r + half * 8;
        int idx = (b * OFFC + ocr) * HW_ + p;
        float v = acc[r] + b_om[ocr];
        offs_ws[idx] = MRM_ * fast_tanh(v) + pre_offset[idx];
      }
    } else if (tile < 13) {              // scalar branch: full mask tiles
      #pragma unroll
      for (int r = 0; r < 8; ++r) {
        int ocr = tile * 16 + r + half * 8;
        int idx = (b * MSKC + (ocr - OFFC)) * HW_ + p;
        float s = (acc[r] + b_om[ocr]) * pre_sim[idx];
        mask_ws[idx] = fast_sigmoid(s);
      }
    } else if (half == 0) {              // tile 13: rows 208..215 only
      #pragma unroll
      for (int r = 0; r < 8; ++r) {
        int ocr = 13 * 16 + r;           // half == 0
        int idx = (b * MSKC + (ocr - OFFC)) * HW_ + p;
        float s = (acc[r] + b_om[ocr]) * pre_sim[idx];
        mask_ws[idx] = fast_sigmoid(s);
      }
    }
    __syncthreads();   // before next tile overwrites wsh
  }
}

// ---------------------------------------------------------------------------
// Kernel 2: modulated deformable conv. One workgroup handles 4 consecutive
// 16-pixel strips of a row: stage the full 64x576 weight matrix in LDS once,
// then per strip: bilinear-gather masked im2col val[576][16] -> 4 waves each
// run one 16(oc) x 16(px) WMMA tile with pure-linear LDS addressing.
// ---------------------------------------------------------------------------
__global__ __launch_bounds__(128) void deform_conv_kernel(
    const float* __restrict__ x0, const float* __restrict__ weight,
    const float* __restrict__ bias, const float* __restrict__ offs_ws,
    const float* __restrict__ mask_ws, float* __restrict__ out)
{
  extern __shared__ float smem[];
  float* wsh = smem;                 // [64][WSTR]
  float* val = smem + K2_WSH;        // [576][16]

  // stage full weight matrix once via b128 (reused for 4 strips)
  #pragma unroll 4
  for (int r = 0; r < O_; ++r) {
    const float4* srcw = reinterpret_cast<const float4*>(weight + (size_t)r * KC1);
    float4* dstw = reinterpret_cast<float4*>(wsh + r * WSTR);
    for (int q = threadIdx.x; q < KC1 / 4; q += 128)
      dstw[q] = srcw[q];
  }
  __syncthreads();

  const int lane = threadIdx.x & 31;
  const int wave = threadIdx.x >> 5;       // oc tile 0..3
  const int m    = lane & 15;
  const int half = lane >> 4;
  const float* arow = wsh + (wave * 16 + m) * WSTR;

  // bias for this lane's 8 D rows, hoisted out of the strip loop
  float bv[8];
  #pragma unroll
  for (int r = 0; r < 8; ++r)
    bv[r] = bias[wave * 16 + r + half * 8];

  for (int si = 0; si < 4; ++si) {
    const int strip = blockIdx.x * 4 + si;       // 2048 strips total
    const int b  = strip / (H_ * 8);
    const int r0 = strip % (H_ * 8);
    const int h  = r0 >> 3;
    const int wb = (r0 & 7) * 16;

    // ---- sampling: (group, tap, pixel) = 8*9*16 = 1152 combos, branchless
    for (int u = threadIdx.x; u < DG_ * KK_ * 16; u += 128) {
      int g  = u / (KK_ * 16);
      int rr = u % (KK_ * 16);
      int t  = rr >> 4;                  // tap: ky=t/3 kx=t%3
      int n  = rr & 15;                  // pixel in strip
      int w  = wb + n;
      int pix = h * W_ + w;

      int occ = (g * KK_ + t) * 2;
      float oy = offs_ws[(b * OFFC + occ    ) * HW_ + pix];
      float ox = offs_ws[(b * OFFC + occ + 1) * HW_ + pix];
      float mk = mask_ws[(b * MSKC + g * KK_ + t) * HW_ + pix];

      float py = oy + (float)(h - 1 + t / 3);
      float px = ox + (float)(w - 1 + t % 3);
      float fy = floorf(py), fx = floorf(px);
      float ly = py - fy,    lx = px - fx;
      float hy = 1.0f - ly,  hx = 1.0f - lx;
      int y0 = (int)fy, x0i = (int)fx;
      int y1 = y0 + 1,  x1i = x0i + 1;

      float iy0 = (y0  >= 0 && y0  < H_) ? 1.0f : 0.0f;
      float iy1 = (y1  >= 0 && y1  < H_) ? 1.0f : 0.0f;
      float ix0 = (x0i >= 0 && x0i < W_) ? 1.0f : 0.0f;
      float ix1 = (x1i >= 0 && x1i < W_) ? 1.0f : 0.0f;
      float w00 = mk * hy * hx * iy0 * ix0;
      float w01 = mk * hy * lx * iy0 * ix1;
      float w10 = mk * ly * hx * iy1 * ix0;
      float w11 = mk * ly * lx * iy1 * ix1;

      int yc0 = min(max(y0, 0), H_ - 1), yc1 = min(max(y1, 0), H_ - 1);
      int xc0 = min(max(x0i, 0), W_ - 1), xc1 = min(max(x1i, 0), W_ - 1);

      const float* xg = x0 + (size_t)((b * C_ + g * CG_) * H_) * W_;
      float* vrow = val + t * 16 + n;
      #pragma unroll
      for (int c = 0; c < CG_; ++c) {
        const float* xp = xg + (size_t)c * HW_;
        float v = w00 * xp[yc0 * W_ + xc0] + w01 * xp[yc0 * W_ + xc1]
                + w10 * xp[yc1 * W_ + xc0] + w11 * xp[yc1 * W_ + xc1];
        vrow[(g * CG_ + c) * KK_ * 16] = v;     // k = (g*8+c)*9 + t
      }
    }
    __syncthreads();

    // ---- GEMM: O=64 -> exactly 4 M-tiles, one per wave
    v8f acc = {};
    #pragma unroll 4
    for (int k4 = 0; k4 < KC1; k4 += 4) {
      const int ka = k4 + half * 2;
      v2f a = *(const v2f*)(arow + ka);
      v2f bm;
      bm.x = val[ka * 16 + m];
      bm.y = val[(ka + 1) * 16 + m];
      acc = __builtin_amdgcn_wmma_f32_16x16x4_f32(
          false, a, false, bm, (short)0, acc, false, false);
    }

    const int p = h * W_ + wb + m;
    #pragma unroll
    for (int r = 0; r < 8; ++r) {
      int oc = wave * 16 + r + half * 8;
      out[(b * O_ + oc) * HW_ + p] = acc[r] + bv[r];
    }
    __syncthreads();   // before next strip overwrites val
  }
}

// ---------------------------------------------------------------------------
extern "C" void kernel_launch(void* const* d_in, const int* in_sizes, int n_in,
                              void* d_out, int out_size, void* d_ws, size_t ws_size,
                              hipStream_t stream) {
  const float* x0         = (const float*)d_in[0];
  const float* x1         = (const float*)d_in[1];
  const float* pre_offset = (const float*)d_in[2];
  const float* pre_sim    = (const float*)d_in[3];
  const float* weight     = (const float*)d_in[4];
  const float* bias       = (const float*)d_in[5];
  const float* w_om       = (const float*)d_in[6];
  const float* b_om       = (const float*)d_in[7];

  float* offs_ws = (float*)d_ws;                                 // 18.9 MB
  float* mask_ws = offs_ws + (size_t)B_ * OFFC * HW_;            //  9.4 MB
  float* w_pad   = mask_ws + (size_t)B_ * MSKC * HW_;            //  0.5 MB
  float* out     = (float*)d_out;

  prep_wpad_kernel<<<dim3((OMP_ * KC1 + 255) / 256), dim3(256), 0, stream>>>(
      w_om, w_pad);
  conv_offset_mask_kernel<<<dim3(B_ * H_ * 2), dim3(128),
                            K1_SMEM * sizeof(float), stream>>>(
      x1, w_pad, b_om, pre_offset, pre_sim, offs_ws, mask_ws);
  deform_conv_kernel<<<dim3(B_ * H_ * 2), dim3(128),
                       K2_SMEM * sizeof(float), stream>>>(
      x0, weight, bias, offs_ws, mask_ws, out);
}